// CGLayer_32040456028687
// MI455X (gfx1250) — compile-verified
//
#include <hip/hip_runtime.h>
#include <hip/hip_bf16.h>

typedef __attribute__((ext_vector_type(2))) float v2f;
typedef __attribute__((ext_vector_type(8))) float v8f;

#define B_     4
#define N_     8192
#define P_     2048
#define C_     128
#define COUT_  256
#define S_     32
#define K_     131
#define KP_    132           // K padded to multiple of 4
#define NPTS   (B_ * P_)     // 8192 query points
#define NTOT   ((float)(B_ * P_ * S_))   // 262144 BN population
#define RAD2   0.64f
#define EPS_   1e-5f
#define WPB    4             // waves (points) per block in gemm kernel

// ---------------------------------------------------------------------------
// 1) shift layer: Conv1d(3->3) + BN(batch stats over B,P) + ReLU  -> new_xyz
// ---------------------------------------------------------------------------
__global__ void shift_kernel(const float* __restrict__ ffps,
                             const float* __restrict__ wsh,
                             const float* __restrict__ gam,
                             const float* __restrict__ bet,
                             float* __restrict__ newxyz) {
  __shared__ float red[6][256];
  __shared__ float sc[3], sh[3];
  const int tid = threadIdx.x;
  float w[9];
  #pragma unroll
  for (int i = 0; i < 9; ++i) w[i] = wsh[i];

  float sum[3] = {0.f, 0.f, 0.f}, sq[3] = {0.f, 0.f, 0.f};
  for (int i = tid; i < NPTS; i += 256) {
    float x0 = ffps[i * 3 + 0], x1 = ffps[i * 3 + 1], x2 = ffps[i * 3 + 2];
    #pragma unroll
    for (int o = 0; o < 3; ++o) {
      float v = w[o * 3 + 0] * x0 + w[o * 3 + 1] * x1 + w[o * 3 + 2] * x2;
      sum[o] += v; sq[o] += v * v;
    }
  }
  #pragma unroll
  for (int o = 0; o < 3; ++o) { red[o][tid] = sum[o]; red[3 + o][tid] = sq[o]; }
  __syncthreads();
  if (tid < 6) {
    float a = 0.f;
    for (int j = 0; j < 256; ++j) a += red[tid][j];
    red[tid][0] = a;
  }
  __syncthreads();
  if (tid < 3) {
    float m = red[tid][0] / (float)NPTS;
    float v = red[3 + tid][0] / (float)NPTS - m * m;
    float s = gam[tid] * rsqrtf(v + EPS_);
    sc[tid] = s; sh[tid] = bet[tid] - m * s;
  }
  __syncthreads();
  for (int i = tid; i < NPTS; i += 256) {
    float x0 = ffps[i * 3 + 0], x1 = ffps[i * 3 + 1], x2 = ffps[i * 3 + 2];
    #pragma unroll
    for (int o = 0; o < 3; ++o) {
      float v = w[o * 3 + 0] * x0 + w[o * 3 + 1] * x1 + w[o * 3 + 2] * x2;
      newxyz[i * 3 + o] = fmaxf(v * sc[o] + sh[o], 0.f);
    }
  }
}

// ---------------------------------------------------------------------------
// 2) ball query: first S_ indices with d2 < RAD2, pad with first hit
//    one wave32 per query point, ballot + ordered bit-scan, early exit
// ---------------------------------------------------------------------------
__global__ void ballquery_kernel(const float* __restrict__ bxyz,
                                 const float* __restrict__ newxyz,
                                 int* __restrict__ idxout) {
  const int lane = threadIdx.x & 31;
  const int pt = blockIdx.x * 8 + (threadIdx.x >> 5);
  const int b = pt / P_;
  const float qx = newxyz[pt * 3 + 0];
  const float qy = newxyz[pt * 3 + 1];
  const float qz = newxyz[pt * 3 + 2];
  const float* base = bxyz + (size_t)b * N_ * 3;

  int cnt = 0, myidx = 0, first = 0;
  for (int j0 = 0; j0 < N_; j0 += 32) {
    const int j = j0 + lane;
    float dx = base[j * 3 + 0] - qx;
    float dy = base[j * 3 + 1] - qy;
    float dz = base[j * 3 + 2] - qz;
    bool in = (dx * dx + dy * dy + dz * dz) < RAD2;
    unsigned mask = (unsigned)__ballot(in);   // wave32: low 32 bits
    while (mask && cnt < S_) {
      int bit = __ffs(mask) - 1;
      mask &= mask - 1;
      int jj = j0 + bit;
      if (cnt == 0) first = jj;
      if (lane == cnt) myidx = jj;
      ++cnt;
    }
    if (cnt >= S_) break;
  }
  if (lane >= cnt) myidx = first;   // pad (cnt==0 -> 0, matching top_k fallback)
  idxout[pt * S_ + lane] = myidx;
}

// ---------------------------------------------------------------------------
// 3) pad w_mlp (COUT x 131 -> COUT x 132, zero last col) + zero stat accums
// ---------------------------------------------------------------------------
__global__ void prep_kernel(const float* __restrict__ wmlp,
                            float* __restrict__ wpad,
                            float* __restrict__ gsum,
                            float* __restrict__ gsq) {
  const int i = blockIdx.x * 256 + threadIdx.x;
  if (i < COUT_ * KP_) {
    int m = i / KP_, k = i % KP_;
    wpad[i] = (k < K_) ? wmlp[m * K_ + k] : 0.f;
  }
  if (i < COUT_) { gsum[i] = 0.f; gsq[i] = 0.f; }
}

// ---------------------------------------------------------------------------
// 4) main pass: gather (32 x 132) tile to LDS, WMMA f32 16x16x4 GEMM vs
//    w_pad (256 x 132), track per-(point,channel) max/min of y over S and
//    per-channel sum/sumsq (LDS -> global atomics). One wave per point.
// ---------------------------------------------------------------------------
__global__ void gemm_kernel(const float* __restrict__ bxyz,
                            const float* __restrict__ feats,
                            const float* __restrict__ newxyz,
                            const int* __restrict__ idx,
                            const float* __restrict__ wpad,
                            float* __restrict__ wsmax,
                            float* __restrict__ wsmin,
                            float* __restrict__ gsum,
                            float* __restrict__ gsq) {
  extern __shared__ float smem[];
  float* gall = smem;                          // WPB * S_ * KP_ floats
  float* ssum = smem + WPB * S_ * KP_;         // COUT_ floats
  float* ssq  = ssum + COUT_;                  // COUT_ floats (contiguous)

  const int tid  = threadIdx.x;
  const int lane = tid & 31;
  const int wv   = tid >> 5;
  const int pt   = blockIdx.x * WPB + wv;
  const int b    = pt / P_;
  float* g = gall + wv * (S_ * KP_);

  // zero block-level stat accumulators (ssum & ssq are contiguous)
  for (int i = tid; i < 2 * COUT_; i += WPB * 32) ssum[i] = 0.f;

  // ---- gather: lane s fills row s of g (row-major [S][KP]) ----
  const int mi = idx[pt * S_ + lane];
  const float cx = newxyz[pt * 3 + 0];
  const float cy = newxyz[pt * 3 + 1];
  const float cz = newxyz[pt * 3 + 2];
  const float* bp = bxyz + ((size_t)b * N_ + mi) * 3;
  g[lane * KP_ + 0] = bp[0] - cx;
  g[lane * KP_ + 1] = bp[1] - cy;
  g[lane * KP_ + 2] = bp[2] - cz;
  const float* fp = feats + (size_t)b * C_ * N_ + mi;
  #pragma unroll 4
  for (int c = 0; c < C_; ++c) g[lane * KP_ + 3 + c] = fp[(size_t)c * N_];
  g[lane * KP_ + K_] = 0.f;   // zero-pad K=131
  __syncthreads();

  // ---- WMMA GEMM: y[256 x 32] = w_pad[256 x 132] * g^T[132 x 32] ----
  // f32 16x16x4 fragments: lanes 0-15 hold K={0,1}, lanes 16-31 hold K={2,3}
  const int half = lane >> 4;        // 0 or 1
  const int l16  = lane & 15;
  const float* arow = wpad + l16 * KP_ + half * 2;          // A: M=l16 row
  const float* b0p  = g + l16 * KP_ + half * 2;             // B tile 0: N=l16
  const float* b1p  = g + (16 + l16) * KP_ + half * 2;      // B tile 1: N=16+l16

  for (int mt = 0; mt < COUT_ / 16; ++mt) {
    v8f acc0 = {};
    v8f acc1 = {};
    const float* am = arow + mt * 16 * KP_;
    for (int k = 0; k < KP_; k += 4) {
      v2f a   = *(const v2f*)(am  + k);   // global_load_b64 (L2-resident w)
      v2f bb0 = *(const v2f*)(b0p + k);   // ds_load_b64
      v2f bb1 = *(const v2f*)(b1p + k);   // ds_load_b64
      acc0 = __builtin_amdgcn_wmma_f32_16x16x4_f32(false, a, false, bb0,
                                                   (short)0, acc0, false, false);
      acc1 = __builtin_amdgcn_wmma_f32_16x16x4_f32(false, a, false, bb1,
                                                   (short)0, acc1, false, false);
    }
    // reduce over the sample (N) axis: VGPR r -> M = mt*16 + half*8 + r
    #pragma unroll
    for (int r = 0; r < 8; ++r) {
      float a0 = acc0[r], a1 = acc1[r];
      float mx = fmaxf(a0, a1);
      float mn = fminf(a0, a1);
      float sm = a0 + a1;
      float sq = a0 * a0 + a1 * a1;
      #pragma unroll
      for (int off = 8; off >= 1; off >>= 1) {   // reduce within 16-lane group
        mx = fmaxf(mx, __shfl_xor(mx, off));
        mn = fminf(mn, __shfl_xor(mn, off));
        sm += __shfl_xor(sm, off);
        sq += __shfl_xor(sq, off);
      }
      if (l16 == 0) {
        int m = mt * 16 + half * 8 + r;
        wsmax[(size_t)pt * COUT_ + m] = mx;
        wsmin[(size_t)pt * COUT_ + m] = mn;
        atomicAdd(&ssum[m], sm);
        atomicAdd(&ssq[m], sq);
      }
    }
  }
  __syncthreads();
  // flush block stats -> global (one atomic per channel per block)
  for (int i = tid; i < COUT_; i += WPB * 32) {
    atomicAdd(&gsum[i], ssum[i]);
    atomicAdd(&gsq[i],  ssq[i]);
  }
}

// ---------------------------------------------------------------------------
// 5) finalize: BN(batch stats) + ReLU + max-over-S via sign-aware max/min
// ---------------------------------------------------------------------------
__global__ void finalize_kernel(const float* __restrict__ wsmax,
                                const float* __restrict__ wsmin,
                                const float* __restrict__ gsum,
                                const float* __restrict__ gsq,
                                const float* __restrict__ gam,
                                const float* __restrict__ bet,
                                float* __restrict__ out) {
  const int i = blockIdx.x * 256 + threadIdx.x;   // over NPTS*COUT
  const int m = i & (COUT_ - 1);
  float mean = gsum[m] * (1.f / NTOT);
  float var  = gsq[m] * (1.f / NTOT) - mean * mean;
  float s    = gam[m] * rsqrtf(var + EPS_);
  float sh   = bet[m] - mean * s;
  float base = (s >= 0.f) ? wsmax[i] : wsmin[i];
  out[i] = fmaxf(base * s + sh, 0.f);
}

// ---------------------------------------------------------------------------
extern "C" void kernel_launch(void* const* d_in, const int* in_sizes, int n_in,
                              void* d_out, int out_size, void* d_ws, size_t ws_size,
                              hipStream_t stream) {
  (void)in_sizes; (void)n_in; (void)out_size; (void)ws_size;
  const float* ffps  = (const float*)d_in[0];   // (B,P,3)
  const float* bxyz  = (const float*)d_in[1];   // (B,N,3)
  const float* feats = (const float*)d_in[2];   // (B,C,N)
  const float* wsh   = (const float*)d_in[3];   // (3,3)
  const float* gsh   = (const float*)d_in[4];   // (3,)
  const float* bsh   = (const float*)d_in[5];   // (3,)
  const float* wmlp  = (const float*)d_in[6];   // (COUT,131)
  const float* gml   = (const float*)d_in[7];   // (COUT,)
  const float* bml   = (const float*)d_in[8];   // (COUT,)
  float* out = (float*)d_out;                   // (B,P,COUT)

  // workspace layout (floats)
  float* ws = (float*)d_ws;
  size_t off = 0;
  float* newxyz = ws + off; off += (size_t)NPTS * 3;        //  98 KB
  int*   idx    = (int*)(ws + off); off += (size_t)NPTS * S_; // 1 MB
  float* wpad   = ws + off; off += (size_t)COUT_ * KP_;     // 135 KB
  float* gsum   = ws + off; off += COUT_;
  float* gsq    = ws + off; off += COUT_;
  float* wsmax  = ws + off; off += (size_t)NPTS * COUT_;    // 8.4 MB
  float* wsmin  = ws + off; off += (size_t)NPTS * COUT_;    // 8.4 MB

  shift_kernel<<<1, 256, 0, stream>>>(ffps, wsh, gsh, bsh, newxyz);
  ballquery_kernel<<<NPTS / 8, 256, 0, stream>>>(bxyz, newxyz, idx);
  prep_kernel<<<(COUT_ * KP_ + 255) / 256, 256, 0, stream>>>(wmlp, wpad, gsum, gsq);

  const size_t shmem = (size_t)(WPB * S_ * KP_ + 2 * COUT_) * sizeof(float); // 69632 B
  gemm_kernel<<<NPTS / WPB, WPB * 32, shmem, stream>>>(
      bxyz, feats, newxyz, idx, wpad, wsmax, wsmin, gsum, gsq);

  finalize_kernel<<<(NPTS * COUT_) / 256, 256, 0, stream>>>(
      wsmax, wsmin, gsum, gsq, gml, bml, out);
}